// GCN_GlobalAttention_78718160601830
// MI455X (gfx1250) — compile-verified
//
#include <hip/hip_runtime.h>
#include <hip/hip_bf16.h>

typedef __attribute__((ext_vector_type(2))) float v2f;
typedef __attribute__((ext_vector_type(8))) float v8f;

#define H 128
#define OUTF 16
#define GNUM 128
#define LLAYERS 3
#define BN_EPS 1e-5f
#define BSTRIDE 68   // padded LDS stride (floats): 8B-aligned, conflict-free reads

// ---------------------------------------------------------------------------
// C[M,128] = A[M,128] @ B[128,128] + bias, optional ReLU.
// Block: 256 threads = 8 waves; block tile = 128 rows x 128 cols.
// Each wave owns a 16-row stripe and 8 accumulator tiles (16x16 each),
// computed with v_wmma_f32_16x16x4_f32 (fp32-exact vs the reference).
// B is staged COLUMN-MAJOR in LDS so each hardware B-fragment
// ({B[k][c], B[k+1][c]} per lane) is one aligned ds_load_b64; fragments for
// a K-step are batched into independent VGPRs so their LDS latency pipelines
// under the WMMA issue instead of serializing on s_wait_dscnt 0.
// ---------------------------------------------------------------------------
__global__ __launch_bounds__(256) void gemm128_wmma(
    const float* __restrict__ A, const float* __restrict__ B,
    const float* __restrict__ bias, float* __restrict__ C,
    int M, int do_relu)
{
  __shared__ float Bs[H][BSTRIDE];  // [col][k-within-half]
  const int tid  = threadIdx.x;
  const int wave = tid >> 5;
  const int lane = tid & 31;
  const int rowBase = blockIdx.x * 128 + wave * 16;
  const int m    = lane & 15;            // row/col within 16-half
  const int koff = (lane >> 4) << 1;     // lanes 0-15 -> K {k,k+1}; 16-31 -> {k+2,k+3}

  v8f acc[8];
#pragma unroll
  for (int j = 0; j < 8; ++j) acc[j] = v8f{};

  // Clamp the A row so loads are always in-bounds; out-of-range rows compute
  // garbage that the guarded epilogue never stores.
  const int rowA = min(rowBase + m, M - 1);
  const float* arowBase = A + (size_t)rowA * H;

  for (int kh = 0; kh < 2; ++kh) {
    // cooperative transpose-stage of one 64xH slab of B into LDS:
    // global read coalesced over columns, LDS write scattered (column-major)
    for (int i = tid; i < 64 * H; i += 256) {
      const int kk = i >> 7;          // 0..63 (k within half)
      const int c  = i & (H - 1);     // 0..127 (column)
      Bs[c][kk] = B[(size_t)(kh * 64 + kk) * H + c];
    }
    __syncthreads();

    const float* arow = arowBase + kh * 64;
    if (kh == 0) __builtin_prefetch(arowBase + 64, 0, 0);  // next K-half

#pragma unroll
    for (int ks = 0; ks < 16; ++ks) {
      const int k = ks * 4;
      const v2f a = *(const v2f*)(arow + k + koff);        // single global b64
      v2f bf[8];
#pragma unroll
      for (int j = 0; j < 8; ++j)                          // 8 independent b64s
        bf[j] = *(const v2f*)(&Bs[j * 16 + m][k + koff]);
#pragma unroll
      for (int j = 0; j < 8; ++j)
        acc[j] = __builtin_amdgcn_wmma_f32_16x16x4_f32(
            false, a, false, bf[j], (short)0, acc[j], false, false);
    }
    __syncthreads();
  }

  // Epilogue: C/D layout = 8 VGPRs; VGPR r: lanes 0-15 -> M=r, lanes 16-31 -> M=r+8
  const int rAdd = (lane >> 4) << 3;
#pragma unroll
  for (int j = 0; j < 8; ++j) {
    const int col = j * 16 + m;
    const float bv = bias[col];
#pragma unroll
    for (int r = 0; r < 8; ++r) {
      const int orow = rowBase + r + rAdd;
      if (orow < M) {
        float v = acc[j][r] + bv;
        if (do_relu) v = fmaxf(v, 0.f);
        C[(size_t)orow * H + col] = v;
      }
    }
  }
}

// ---------------------------------------------------------------------------
__global__ void zero_kernel(float* __restrict__ p, size_t n) {
  size_t i = (size_t)blockIdx.x * blockDim.x + threadIdx.x;
  if (i < n) p[i] = 0.f;
}

// One wave per edge: lane handles a float4 feature chunk, 4 fp32 atomics.
__global__ __launch_bounds__(256) void scatter_edges(
    const int* __restrict__ rows, const int* __restrict__ cols,
    const float* __restrict__ w, const float* __restrict__ hn,
    float* __restrict__ agg, int E)
{
  const int e = blockIdx.x * 8 + (threadIdx.x >> 5);
  if (e >= E) return;
  const int lane = threadIdx.x & 31;
  const int r = rows[e], c = cols[e];
  const float ww = w[e];
  const float4 v = ((const float4*)(hn + (size_t)c * H))[lane];
  float* dst = agg + (size_t)r * H + lane * 4;
  atomicAdd(dst + 0, ww * v.x);
  atomicAdd(dst + 1, ww * v.y);
  atomicAdd(dst + 2, ww * v.z);
  atomicAdd(dst + 3, ww * v.w);
}

// h[i] += relu( (agg[i]-mean[c]) * gamma[c]*rsqrt(var[c]+eps) + beta[c] )
__global__ void bn_relu_residual(
    const float* __restrict__ agg,
    const float* __restrict__ gamma, const float* __restrict__ beta,
    const float* __restrict__ mean, const float* __restrict__ var,
    float* __restrict__ h, int total)
{
  const int i = blockIdx.x * blockDim.x + threadIdx.x;
  if (i >= total) return;
  const int c = i & (H - 1);
  const float scale = gamma[c] * rsqrtf(var[c] + BN_EPS);
  const float v = (agg[i] - mean[c]) * scale + beta[c];
  h[i] += fmaxf(v, 0.f);
}

// One wave per node: g[n] = <h[n], gate_W> + gate_b  (wave32 shuffle reduce)
__global__ __launch_bounds__(256) void gate_kernel(
    const float* __restrict__ h, const float* __restrict__ gw,
    const float* __restrict__ gb, float* __restrict__ g, int Nn)
{
  const int n = blockIdx.x * 8 + (threadIdx.x >> 5);
  if (n >= Nn) return;
  const int lane = threadIdx.x & 31;
  const float4 hv = ((const float4*)(h + (size_t)n * H))[lane];
  const float4 wv = ((const float4*)gw)[lane];
  float s = hv.x * wv.x + hv.y * wv.y + hv.z * wv.z + hv.w * wv.w;
#pragma unroll
  for (int off = 16; off > 0; off >>= 1) s += __shfl_xor(s, off, 32);
  if (lane == 0) g[n] = s + gb[0];
}

// One block per graph: binary-search node range in sorted batch[], then
// segment max -> exp-sum -> weighted feature accumulation (thread = feature).
__global__ __launch_bounds__(128) void pool_kernel(
    const float* __restrict__ h, const float* __restrict__ gv,
    const int* __restrict__ batch, float* __restrict__ emb, int Nn)
{
  __shared__ float red[128];
  __shared__ float s_m, s_s;
  const int gid = blockIdx.x;
  const int tid = threadIdx.x;

  int lo = 0, hi = Nn;
  while (lo < hi) { int mid = (lo + hi) >> 1; if (batch[mid] < gid) lo = mid + 1; else hi = mid; }
  const int start = lo;
  hi = Nn;
  while (lo < hi) { int mid = (lo + hi) >> 1; if (batch[mid] < gid + 1) lo = mid + 1; else hi = mid; }
  const int end = lo;

  if (end <= start) { emb[(size_t)gid * H + tid] = 0.f; return; }

  // segment max
  float lm = -3.402823466e38f;
  for (int n = start + tid; n < end; n += 128) lm = fmaxf(lm, gv[n]);
  red[tid] = lm; __syncthreads();
  for (int off = 64; off > 0; off >>= 1) {
    if (tid < off) red[tid] = fmaxf(red[tid], red[tid + off]);
    __syncthreads();
  }
  if (tid == 0) s_m = red[0];
  __syncthreads();
  const float mval = s_m;

  // sum of exp
  float ls = 0.f;
  for (int n = start + tid; n < end; n += 128) ls += expf(gv[n] - mval);
  red[tid] = ls; __syncthreads();
  for (int off = 64; off > 0; off >>= 1) {
    if (tid < off) red[tid] += red[tid + off];
    __syncthreads();
  }
  if (tid == 0) s_s = red[0];
  __syncthreads();
  const float denom = s_s + 1e-10f;

  // weighted pooling: thread tid owns feature channel tid (coalesced reads)
  float acc = 0.f;
  for (int n = start; n < end; ++n)
    acc += h[(size_t)n * H + tid] * expf(gv[n] - mval);
  emb[(size_t)gid * H + tid] = acc / denom;
}

// out[g, o] = sum_k t1[g,k] * W2[k,o] + b2[o]   (128 x 16, trivial)
__global__ void head2_kernel(const float* __restrict__ t1,
    const float* __restrict__ W2, const float* __restrict__ b2,
    float* __restrict__ out)
{
  const int id = blockIdx.x * blockDim.x + threadIdx.x;
  if (id >= GNUM * OUTF) return;
  const int gi = id / OUTF, o = id % OUTF;
  float acc = b2[o];
  for (int k = 0; k < H; ++k) acc += t1[(size_t)gi * H + k] * W2[k * OUTF + o];
  out[id] = acc;
}

// ---------------------------------------------------------------------------
extern "C" void kernel_launch(void* const* d_in, const int* in_sizes, int n_in,
                              void* d_out, int out_size, void* d_ws, size_t ws_size,
                              hipStream_t stream)
{
  const float* x       = (const float*)d_in[0];
  const float* edge_w  = (const float*)d_in[1];
  const float* W_in    = (const float*)d_in[2];
  const float* b_in    = (const float*)d_in[3];
  const float* conv_W  = (const float*)d_in[4];
  const float* conv_b  = (const float*)d_in[5];
  const float* bn_g    = (const float*)d_in[6];
  const float* bn_b    = (const float*)d_in[7];
  const float* bn_m    = (const float*)d_in[8];
  const float* bn_v    = (const float*)d_in[9];
  const float* gate_W  = (const float*)d_in[10];
  const float* gate_b  = (const float*)d_in[11];
  const float* head_W1 = (const float*)d_in[12];
  const float* head_b1 = (const float*)d_in[13];
  const float* head_W2 = (const float*)d_in[14];
  const float* head_b2 = (const float*)d_in[15];
  const int*   erows   = (const int*)d_in[16];
  const int*   ecols   = (const int*)d_in[17];
  const int*   batch   = (const int*)d_in[18];

  const int N = in_sizes[18];   // nodes
  const int E = in_sizes[16];   // edges
  const size_t NH = (size_t)N * H;

  float* ws  = (float*)d_ws;
  float* h   = ws;                       // [N,H]
  float* hn  = h   + NH;                 // [N,H]
  float* agg = hn  + NH;                 // [N,H]
  float* gv  = agg + NH;                 // [N]
  float* emb = gv  + N;                  // [G,H]
  float* t1  = emb + (size_t)GNUM * H;   // [G,H]
  float* out = (float*)d_out;            // [G,OUT]

  const int gemmGrid = (N + 127) / 128;
  const int ewGrid   = (int)((NH + 255) / 256);

  // h = relu(x @ W_in + b_in)
  gemm128_wmma<<<gemmGrid, 256, 0, stream>>>(x, W_in, b_in, h, N, 1);

  for (int l = 0; l < LLAYERS; ++l) {
    gemm128_wmma<<<gemmGrid, 256, 0, stream>>>(
        h, conv_W + (size_t)l * H * H, conv_b + (size_t)l * H, hn, N, 0);
    zero_kernel<<<ewGrid, 256, 0, stream>>>(agg, NH);
    scatter_edges<<<(E + 7) / 8, 256, 0, stream>>>(erows, ecols, edge_w, hn, agg, E);
    bn_relu_residual<<<ewGrid, 256, 0, stream>>>(
        agg, bn_g + (size_t)l * H, bn_b + (size_t)l * H,
        bn_m + (size_t)l * H, bn_v + (size_t)l * H, h, (int)NH);
  }

  gate_kernel<<<(N + 7) / 8, 256, 0, stream>>>(h, gate_W, gate_b, gv, N);
  pool_kernel<<<GNUM, 128, 0, stream>>>(h, gv, batch, emb, N);

  // head: t1 = relu(emb @ head_W1 + head_b1)  (single 128-row block, WMMA)
  gemm128_wmma<<<1, 256, 0, stream>>>(emb, head_W1, head_b1, t1, GNUM, 1);
  head2_kernel<<<8, 256, 0, stream>>>(t1, head_W2, head_b2, out);
}